// GAD_Explainer_44100724195779
// MI455X (gfx1250) — compile-verified
//
#include <hip/hip_runtime.h>
#include <math.h>

#ifndef __has_builtin
#define __has_builtin(x) 0
#endif

// ---- problem sizes (fixed by reference) ----
#define NN  20000
#define EE  320000
#define GG  256
#define DIN 128
#define DD  256
#define PP  16
#define EPSC 1e-7f
#define RRC  0.5f
#define TEMPC 0.2f

#define KMAX 512           // largest GEMM K (mask MLP)
#define WPB  8             // waves per GEMM block

typedef float v2f __attribute__((ext_vector_type(2)));
typedef float v8f __attribute__((ext_vector_type(8)));
typedef int   v4i __attribute__((vector_size(4 * sizeof(int))));

#define AS1 __attribute__((address_space(1)))
#define AS3 __attribute__((address_space(3)))
typedef AS1 v4i* g_v4i_p;   // global 128-bit chunk pointer
typedef AS3 v4i* l_v4i_p;   // LDS 128-bit chunk pointer

static __device__ __forceinline__ void atomAdd(float* p, float v) {
  unsafeAtomicAdd(p, v);   // lowers to global_atomic_add_f32 on gfx1250
}

static __device__ __forceinline__ void wait_async0() {
#if __has_builtin(__builtin_amdgcn_s_wait_asynccnt)
  __builtin_amdgcn_s_wait_asynccnt(0);
#else
  asm volatile("s_wait_asynccnt 0x0" ::: "memory");
#endif
}

// ---------------- elementwise / memory kernels ----------------

__global__ void k_zero(float* __restrict__ p, size_t n) {
  size_t i = (size_t)blockIdx.x * blockDim.x + threadIdx.x;
  if (i < n) p[i] = 0.f;
}

__global__ void k_copy(const float* __restrict__ a, float* __restrict__ b, size_t n) {
  size_t i = (size_t)blockIdx.x * blockDim.x + threadIdx.x;
  if (i < n) b[i] = a[i];
}

// agg[dst[e], f] += h[src[e], f] * (att ? att[e] : 1)
__global__ void k_scatter(const float* __restrict__ h, const int* __restrict__ src,
                          const int* __restrict__ dst, const float* __restrict__ att,
                          float* __restrict__ agg, int D) {
  size_t i = (size_t)blockIdx.x * blockDim.x + threadIdx.x;
  size_t tot = (size_t)EE * (size_t)D;
  if (i >= tot) return;
  int e = (int)(i / (size_t)D);
  int f = (int)(i % (size_t)D);
  float v = h[(size_t)src[e] * D + f];
  if (att) v *= att[e];
  atomAdd(&agg[(size_t)dst[e] * D + f], v);
}

// gout[batch[n], f] += h[n, f]
__global__ void k_segsum(const float* __restrict__ h, const int* __restrict__ batch,
                         float* __restrict__ gout) {
  size_t i = (size_t)blockIdx.x * blockDim.x + threadIdx.x;
  if (i >= (size_t)NN * DD) return;
  int n = (int)(i / DD);
  int f = (int)(i % DD);
  atomAdd(&gout[(size_t)batch[n] * DD + f], h[i]);
}

// ---------------- WMMA fp32 GEMM: C = relu(A[MxK] @ W[KxNc] + b) ----------------
// grid = (ceil(M/16/WPB), Nc/16); block = WPB waves.
// The K x 16 weight panel for this block's column strip is staged once into LDS
// via GLOBAL_LOAD_ASYNC_TO_LDS_B128 (ASYNCcnt), then each wave computes one
// 16x16 C tile with V_WMMA_F32_16X16X4_F32, B-fragments fed from ds_load.
__global__ void k_gemm_relu(const float* __restrict__ A, const float* __restrict__ W,
                            const float* __restrict__ b, float* __restrict__ C,
                            int M, int K, int Nc) {
  __shared__ float ldsW[KMAX * 16];          // 32 KB max panel

  int tn = blockIdx.y;                        // column strip (16 cols)
  int tilesM = M >> 4;

  // ---- cooperative async copy: W[0:K, tn*16 : tn*16+16] -> ldsW (row-major K x 16)
  {
    const float* gbase = W + (size_t)tn * 16;
    int nch = K * 4;                          // number of float4 chunks in panel
    for (int c = threadIdx.x; c < nch; c += blockDim.x) {
      int row = c >> 2;                       // panel row (K index)
      int cb  = (c & 3) * 4;                  // column sub-block (float4)
      const float* gp = gbase + (size_t)row * Nc + cb;
      float* lp = &ldsW[(size_t)c * 4];
#if __has_builtin(__builtin_amdgcn_global_load_async_to_lds_b128)
      __builtin_amdgcn_global_load_async_to_lds_b128(
          (g_v4i_p)gp, (l_v4i_p)lp, 0, 0);
#else
      lp[0] = gp[0]; lp[1] = gp[1]; lp[2] = gp[2]; lp[3] = gp[3];
#endif
    }
#if __has_builtin(__builtin_amdgcn_global_load_async_to_lds_b128)
    wait_async0();
#endif
  }
  __syncthreads();

  int wave = threadIdx.x >> 5;
  int tm   = blockIdx.x * WPB + wave;
  int lane = threadIdx.x & 31;
  int half = lane >> 4;                       // 0: lanes 0-15, 1: lanes 16-31
  int m    = lane & 15;

  if (tm < tilesM) {                          // wave-uniform: EXEC all-1s for WMMA
    const float* Arow = A + (size_t)(tm * 16 + m) * K;
    v8f acc = {};
    for (int kk = 0; kk < K; kk += 4) {
      if (kk + 32 < K) __builtin_prefetch(Arow + kk + 32, 0, 1);
      int ka = kk + 2 * half;
      v2f av; av.x = Arow[ka];            av.y = Arow[ka + 1];            // A[m][ka+v]
      v2f bv; bv.x = ldsW[ka * 16 + m];   bv.y = ldsW[(ka + 1) * 16 + m]; // B[ka+v][m]
      acc = __builtin_amdgcn_wmma_f32_16x16x4_f32(
          /*neg_a=*/false, av, /*neg_b=*/false, bv,
          /*c_mod=*/(short)0, acc, /*reuse_a=*/false, /*reuse_b=*/false);
    }
    int col = tn * 16 + m;
    float bias = b[col];
#pragma unroll
    for (int r = 0; r < 8; ++r) {
      int row = tm * 16 + r + 8 * half;
      float v = acc[r] + bias;
      v = fmaxf(v, 0.f);
      C[(size_t)row * Nc + col] = v;
    }
  }
}

// ---------------- small dense epilogue kernels ----------------

// out[r] = ||A[r,:DD]||, zero -> EPS
__global__ void k_rownorm(const float* __restrict__ A, float* __restrict__ out, int rows) {
  int r = blockIdx.x * blockDim.x + threadIdx.x;
  if (r >= rows) return;
  float s = 0.f;
  for (int k = 0; k < DD; ++k) { float v = A[(size_t)r * DD + k]; s += v * v; }
  float nrm = sqrtf(s);
  if (nrm == 0.f) nrm = EPSC;
  out[r] = nrm;
}

// out[a,b] = <A[a],B[b]> / (na[a]*nb[b])
__global__ void k_cossim(const float* __restrict__ A, const float* __restrict__ B,
                         const float* __restrict__ na, const float* __restrict__ nb,
                         float* __restrict__ out, int RA, int RB) {
  int i = blockIdx.x * blockDim.x + threadIdx.x;
  if (i >= RA * RB) return;
  int a = i / RB, bb = i % RB;
  float s = 0.f;
  for (int k = 0; k < DD; ++k) s += A[(size_t)a * DD + k] * B[(size_t)bb * DD + k];
  out[i] = s / (na[a] * nb[bb]);
}

__global__ void k_argmax(const float* __restrict__ sim0, int* __restrict__ asn) {
  int g = blockIdx.x * blockDim.x + threadIdx.x;
  if (g >= GG) return;
  float mx = sim0[(size_t)g * PP];
  int am = 0;
  for (int p = 1; p < PP; ++p) {
    float v = sim0[(size_t)g * PP + p];
    if (v > mx) { mx = v; am = p; }
  }
  asn[g] = am;
}

// hc[n, 0:256] = node_embs[n], hc[n, 256:512] = prototypes[asn[batch[n]]]
__global__ void k_hcat(const float* __restrict__ ne, const float* __restrict__ proto,
                       const int* __restrict__ asn, const int* __restrict__ batch,
                       float* __restrict__ hc) {
  size_t i = (size_t)blockIdx.x * blockDim.x + threadIdx.x;
  if (i >= (size_t)NN * 512) return;
  int n = (int)(i / 512);
  int f = (int)(i % 512);
  hc[i] = (f < DD) ? ne[(size_t)n * DD + f]
                   : proto[(size_t)asn[batch[n]] * DD + (f - DD)];
}

// nb[n] = sigmoid(hidden[n] . Wm2 + bm2)
__global__ void k_nodeprob(const float* __restrict__ hid, const float* __restrict__ Wm2,
                           const float* __restrict__ bm2, float* __restrict__ nb) {
  int n = blockIdx.x * blockDim.x + threadIdx.x;
  if (n >= NN) return;
  float s = bm2[0];
  for (int k = 0; k < DD; ++k) s += hid[(size_t)n * DD + k] * Wm2[k];
  nb[n] = 1.f / (1.f + expf(-s));
}

__global__ void k_edgebern(const float* __restrict__ nb, const int* __restrict__ src,
                           const int* __restrict__ dst, float* __restrict__ eb) {
  int e = blockIdx.x * blockDim.x + threadIdx.x;
  if (e >= EE) return;
  eb[e] = nb[src[e]] * nb[dst[e]];
}

__global__ void k_x2(const float* __restrict__ x, const float* __restrict__ nb,
                     float* __restrict__ x2) {
  size_t i = (size_t)blockIdx.x * blockDim.x + threadIdx.x;
  if (i >= (size_t)NN * DIN) return;
  x2[i] = x[i] * nb[i / DIN];
}

// single 256-thread block: NCE = -mean_g log(pos/neg)
__global__ void k_nce(const float* __restrict__ sim, float* __restrict__ out_nce) {
  __shared__ float red[GG];
  int g = threadIdx.x;
  float mx = sim[(size_t)g * PP];
  int am = 0;
  for (int p = 1; p < PP; ++p) {
    float v = sim[(size_t)g * PP + p];
    if (v > mx) { mx = v; am = p; }
  }
  float pos = 0.f, neg = 0.f;
  for (int p = 0; p < PP; ++p) {
    float e = expf(sim[(size_t)g * PP + p] / TEMPC);
    if (p == am) pos = e; else neg += e;
  }
  red[g] = -logf(pos / neg);
  __syncthreads();
  for (int off = GG / 2; off > 0; off >>= 1) {
    if (g < off) red[g] += red[g + off];
    __syncthreads();
  }
  if (g == 0) out_nce[0] = red[0] / (float)GG;
}

static __device__ __forceinline__ float warp_sum(float v) {
  for (int off = 16; off > 0; off >>= 1) v += __shfl_down(v, off, 32);
  return v;
}

__global__ void k_kl_node(const float* __restrict__ nb, float* __restrict__ acc) {
  size_t i = (size_t)blockIdx.x * blockDim.x + threadIdx.x;
  float t = 0.f;
  for (size_t j = i; j < (size_t)NN; j += (size_t)gridDim.x * blockDim.x) {
    float p = nb[j];
    t += p * logf(p / RRC + EPSC)
       + (1.f - p) * logf((1.f - p) / (1.f - RRC + EPSC) + EPSC);
  }
  t = warp_sum(t);
  if ((threadIdx.x & 31) == 0) atomAdd(acc, t);
}

__global__ void k_kl_edge(const float* __restrict__ eb, float* __restrict__ acc) {
  size_t i = (size_t)blockIdx.x * blockDim.x + threadIdx.x;
  const float rr = RRC * RRC;
  float t = 0.f;
  for (size_t j = i; j < (size_t)EE; j += (size_t)gridDim.x * blockDim.x) {
    float p = eb[j];
    t += p * logf(p / rr + EPSC)
       + (1.f - p) * logf((1.f - p) / (1.f - rr + EPSC) + EPSC);
  }
  t = warp_sum(t);
  if ((threadIdx.x & 31) == 0) atomAdd(acc, t);
}

__global__ void k_final(const float* __restrict__ acc, float* __restrict__ out_kl) {
  if (threadIdx.x == 0 && blockIdx.x == 0)
    out_kl[0] = acc[0] / (float)NN + acc[1] / (float)EE;
}

__global__ void k_datasim(const float* __restrict__ S, const float* __restrict__ sn,
                          float* __restrict__ out) {
  int i = blockIdx.x * blockDim.x + threadIdx.x;
  if (i >= GG * GG) return;
  int a = i / GG, b = i % GG;
  float s = 0.f;
  for (int k = 0; k < DD; ++k) s += S[(size_t)a * DD + k] * S[(size_t)b * DD + k];
  out[i] = s / (sn[a] * sn[b]);
}

// ---------------- host launcher ----------------

static inline unsigned cdiv(size_t n, unsigned b) { return (unsigned)((n + b - 1) / b); }

extern "C" void kernel_launch(void* const* d_in, const int* in_sizes, int n_in,
                              void* d_out, int out_size, void* d_ws, size_t ws_size,
                              hipStream_t stream) {
  (void)in_sizes; (void)n_in; (void)out_size; (void)ws_size;
  const float* x     = (const float*)d_in[0];
  const int*   eidx  = (const int*)d_in[1];
  const int*   src   = eidx;
  const int*   dst   = eidx + EE;
  const int*   batch = (const int*)d_in[2];
  const float* W1    = (const float*)d_in[3];
  const float* b1    = (const float*)d_in[4];
  const float* W2    = (const float*)d_in[5];
  const float* b2    = (const float*)d_in[6];
  const float* Wm1   = (const float*)d_in[7];
  const float* bm1   = (const float*)d_in[8];
  const float* Wm2   = (const float*)d_in[9];
  const float* bm2   = (const float*)d_in[10];
  const float* proto = (const float*)d_in[11];

  float* out = (float*)d_out;
  // output layout (flat, reference return order)
  const size_t O_KL = 0, O_NCE = 1, O_SIM = 2;
  const size_t O_NB = O_SIM + (size_t)GG * PP;
  const size_t O_EB = O_NB + NN;
  const size_t O_DS = O_EB + EE;
  const size_t O_SE = O_DS + (size_t)GG * GG;
  const size_t O_NE = O_SE + (size_t)GG * DD;

  // workspace layout (floats)
  float* ws = (float*)d_ws;
  float* bufA = ws;                                 // N*512 (agg1 / hcat / agg1' / h2')
  float* bufB = bufA + (size_t)NN * 512;            // N*256 (h1 / hidden / h1')
  float* bufC = bufB + (size_t)NN * 256;            // N*256 (agg2 / x2 / agg2')
  float* gemb = bufC + (size_t)NN * 256;            // G*D
  float* gn   = gemb + (size_t)GG * DD;             // G
  float* pn   = gn + GG;                            // P
  float* sn   = pn + PP;                            // G
  float* sim0 = sn + GG;                            // G*P
  int*   asn  = (int*)(sim0 + (size_t)GG * PP);     // G ints
  float* acc  = (float*)(asn + GG);                 // 2

  const unsigned T = 256;
  const dim3 gemmGrid(cdiv(NN / 16, WPB), DD / 16); // 157 x 16 blocks, 8 waves each
  const dim3 gemmBlk(32 * WPB);

  // ---------------- encoder pass 1 ----------------
  hipLaunchKernelGGL(k_copy, dim3(cdiv((size_t)NN * DIN, T)), dim3(T), 0, stream, x, bufA, (size_t)NN * DIN);
  hipLaunchKernelGGL(k_scatter, dim3(cdiv((size_t)EE * DIN, T)), dim3(T), 0, stream, x, src, dst, (const float*)nullptr, bufA, DIN);
  hipLaunchKernelGGL(k_gemm_relu, gemmGrid, gemmBlk, 0, stream, bufA, W1, b1, bufB, NN, DIN, DD);   // h1

  hipLaunchKernelGGL(k_copy, dim3(cdiv((size_t)NN * DD, T)), dim3(T), 0, stream, bufB, bufC, (size_t)NN * DD);
  hipLaunchKernelGGL(k_scatter, dim3(cdiv((size_t)EE * DD, T)), dim3(T), 0, stream, bufB, src, dst, (const float*)nullptr, bufC, DD);
  hipLaunchKernelGGL(k_gemm_relu, gemmGrid, gemmBlk, 0, stream, bufC, W2, b2, out + O_NE, NN, DD, DD);  // node_embs

  hipLaunchKernelGGL(k_zero, dim3(cdiv((size_t)GG * DD, T)), dim3(T), 0, stream, gemb, (size_t)GG * DD);
  hipLaunchKernelGGL(k_segsum, dim3(cdiv((size_t)NN * DD, T)), dim3(T), 0, stream, out + O_NE, batch, gemb);

  hipLaunchKernelGGL(k_rownorm, dim3(cdiv(GG, T)), dim3(T), 0, stream, gemb, gn, GG);
  hipLaunchKernelGGL(k_rownorm, dim3(1), dim3(PP), 0, stream, proto, pn, PP);
  hipLaunchKernelGGL(k_cossim, dim3(cdiv((size_t)GG * PP, T)), dim3(T), 0, stream, gemb, proto, gn, pn, sim0, GG, PP);
  hipLaunchKernelGGL(k_argmax, dim3(cdiv(GG, T)), dim3(T), 0, stream, sim0, asn);

  // ---------------- mask MLP ----------------
  hipLaunchKernelGGL(k_hcat, dim3(cdiv((size_t)NN * 512, T)), dim3(T), 0, stream, out + O_NE, proto, asn, batch, bufA);
  hipLaunchKernelGGL(k_gemm_relu, gemmGrid, gemmBlk, 0, stream, bufA, Wm1, bm1, bufB, NN, 512, DD);  // hidden
  hipLaunchKernelGGL(k_nodeprob, dim3(cdiv(NN, T)), dim3(T), 0, stream, bufB, Wm2, bm2, out + O_NB);
  hipLaunchKernelGGL(k_edgebern, dim3(cdiv(EE, T)), dim3(T), 0, stream, out + O_NB, src, dst, out + O_EB);

  // ---------------- encoder pass 2 (attenuated) ----------------
  hipLaunchKernelGGL(k_x2, dim3(cdiv((size_t)NN * DIN, T)), dim3(T), 0, stream, x, out + O_NB, bufC);
  hipLaunchKernelGGL(k_copy, dim3(cdiv((size_t)NN * DIN, T)), dim3(T), 0, stream, bufC, bufA, (size_t)NN * DIN);
  hipLaunchKernelGGL(k_scatter, dim3(cdiv((size_t)EE * DIN, T)), dim3(T), 0, stream, bufC, src, dst, out + O_EB, bufA, DIN);
  hipLaunchKernelGGL(k_gemm_relu, gemmGrid, gemmBlk, 0, stream, bufA, W1, b1, bufB, NN, DIN, DD);   // h1'

  hipLaunchKernelGGL(k_copy, dim3(cdiv((size_t)NN * DD, T)), dim3(T), 0, stream, bufB, bufC, (size_t)NN * DD);
  hipLaunchKernelGGL(k_scatter, dim3(cdiv((size_t)EE * DD, T)), dim3(T), 0, stream, bufB, src, dst, out + O_EB, bufC, DD);
  hipLaunchKernelGGL(k_gemm_relu, gemmGrid, gemmBlk, 0, stream, bufC, W2, b2, bufA, NN, DD, DD);    // h2' (bufA reuse)

  hipLaunchKernelGGL(k_zero, dim3(cdiv((size_t)GG * DD, T)), dim3(T), 0, stream, out + O_SE, (size_t)GG * DD);
  hipLaunchKernelGGL(k_segsum, dim3(cdiv((size_t)NN * DD, T)), dim3(T), 0, stream, bufA, batch, out + O_SE);

  // ---------------- epilogues ----------------
  hipLaunchKernelGGL(k_rownorm, dim3(cdiv(GG, T)), dim3(T), 0, stream, out + O_SE, sn, GG);
  hipLaunchKernelGGL(k_cossim, dim3(cdiv((size_t)GG * PP, T)), dim3(T), 0, stream, out + O_SE, proto, sn, pn, out + O_SIM, GG, PP);
  hipLaunchKernelGGL(k_nce, dim3(1), dim3(GG), 0, stream, out + O_SIM, out + O_NCE);

  hipLaunchKernelGGL(k_zero, dim3(1), dim3(2), 0, stream, acc, (size_t)2);
  hipLaunchKernelGGL(k_kl_node, dim3(80), dim3(T), 0, stream, out + O_NB, acc + 0);
  hipLaunchKernelGGL(k_kl_edge, dim3(320), dim3(T), 0, stream, out + O_EB, acc + 1);
  hipLaunchKernelGGL(k_final, dim3(1), dim3(1), 0, stream, acc, out + O_KL);

  hipLaunchKernelGGL(k_datasim, dim3(cdiv((size_t)GG * GG, T)), dim3(T), 0, stream, out + O_SE, sn, out + O_DS);
}